// SpatialAttention_75608604279340
// MI455X (gfx1250) — compile-verified
//
#include <hip/hip_runtime.h>
#include <math.h>

typedef float v2f __attribute__((ext_vector_type(2)));
typedef float v8f __attribute__((ext_vector_type(8)));

#define PI_F 3.14159265358979323846f

// Problem / padded sizes
#define CDIM 271
#define CPAD 320      // col padding (multiple of 64)
#define KPW  272      // proj_w K padding (multiple of 16)
#define ODIM 512
#define SDIM 512
#define BDIM 64
#define LDA_S 20      // LDS A-slab row stride in floats (16B-aligned, bank-friendly)

// ---------------------------------------------------------------------------
// Stage 1a: reduce fourier_features [O,32,32,4] -> F[O,128]
// ---------------------------------------------------------------------------
__global__ __launch_bounds__(128) void reduce_fourier_kernel(
    const float* __restrict__ ff, float* __restrict__ F) {
  const int o = blockIdx.x;      // 512
  const int t = threadIdx.x;     // 128
  const int seg = t >> 5, j = t & 31;
  const float* base = ff + (size_t)o * 32 * 32 * 4;
  float s = 0.f;
  if (seg == 0) {
#pragma unroll 4
    for (int w = 0; w < 32; ++w) s += base[(j * 32 + w) * 4 + 0];
  } else if (seg == 1) {
#pragma unroll 4
    for (int h = 0; h < 32; ++h) s += base[(h * 32 + j) * 4 + 1];
  } else if (seg == 2) {
#pragma unroll 4
    for (int w = 0; w < 32; ++w) s += base[(j * 32 + w) * 4 + 2];
  } else {
#pragma unroll 4
    for (int h = 0; h < 32; ++h) s += base[(h * 32 + j) * 4 + 3];
  }
  F[o * 128 + t] = s;
}

// ---------------------------------------------------------------------------
// Stage 1b: S[128, CPAD]: sin/cos basis; columns >= C are zero.
// ---------------------------------------------------------------------------
__global__ __launch_bounds__(128) void build_smat_kernel(
    const float* __restrict__ pos, float* __restrict__ S) {
  const int c = blockIdx.x;      // CPAD
  const int k = threadIdx.x;     // 128
  const int seg = k >> 5, j = k & 31;
  float v = 0.f;
  if (c < CDIM) {
    const float f = 2.f * PI_F * (float)j / 32.f;
    const float px = pos[c * 2 + 0];
    const float py = pos[c * 2 + 1];
    if (seg == 0)      v = sinf(px * f);
    else if (seg == 1) v = sinf(py * f);
    else if (seg == 2) v = cosf(px * f);
    else               v = cosf(py * f);
  }
  S[k * CPAD + c] = v;
}

// ---------------------------------------------------------------------------
// proj_w [512,271] -> padded [512,272] (last column zero)
// ---------------------------------------------------------------------------
__global__ __launch_bounds__(256) void pad_projw_kernel(
    const float* __restrict__ W, float* __restrict__ Wp) {
  const int idx = blockIdx.x * 256 + threadIdx.x;
  if (idx >= ODIM * KPW) return;
  const int r = idx / KPW, c = idx % KPW;
  Wp[idx] = (c < CDIM) ? W[r * CDIM + c] : 0.f;
}

// ---------------------------------------------------------------------------
// x [64,271,512] -> xp [64,320,512], rows >= 271 zero. float4 copies.
// ---------------------------------------------------------------------------
__global__ __launch_bounds__(256) void pad_x_kernel(
    const float* __restrict__ x, float* __restrict__ xp) {
  const size_t total = (size_t)BDIM * CPAD * (SDIM / 4);
  const size_t idx = (size_t)blockIdx.x * 256 + threadIdx.x;
  if (idx >= total) return;
  const int q = (int)(idx % (SDIM / 4));
  const size_t row = idx / (SDIM / 4);       // b*CPAD + r
  const int r = (int)(row % CPAD);
  const int b = (int)(row / CPAD);
  float4 v = make_float4(0.f, 0.f, 0.f, 0.f);
  if (r < CDIM) v = *(const float4*)(x + ((size_t)b * CDIM + r) * SDIM + q * 4);
  *(float4*)(xp + row * SDIM + q * 4) = v;
}

// ---------------------------------------------------------------------------
// Async DMA: one 16B global->LDS transfer per lane (ASYNCcnt-tracked).
// ---------------------------------------------------------------------------
__device__ __forceinline__ void async_b128(uint32_t lds_addr, const float* gaddr) {
  asm volatile("global_load_async_to_lds_b128 %0, %1, off"
               :: "v"(lds_addr), "v"(gaddr) : "memory");
}

// ---------------------------------------------------------------------------
// Fully-padded f32 WMMA GEMM with double-buffered async LDS staging.
// Block = 128 threads = 4 waves (2x2); wave tile 32x32 (2x2 WMMA frags);
// block tile 64x64. K-slab = 16 (multiple of 16 required). All operand
// buffers padded: no guards in the K loop. Stores bounds-checked.
// ---------------------------------------------------------------------------
__device__ __forceinline__ void store_c_frag(float* __restrict__ D, int ldc,
                                             int Mst, int Nst, int row0, int col0,
                                             v8f acc, int lane,
                                             const float* __restrict__ bias) {
  const int n = col0 + (lane & 15);
  const int rbase = row0 + ((lane >> 4) << 3);  // lanes 16-31 hold rows +8..+15
  if (n < Nst) {
#pragma unroll
    for (int v = 0; v < 8; ++v) {
      const int r = rbase + v;
      if (r < Mst) {
        float val = acc[v];
        if (bias) val += bias[r];
        D[(size_t)r * ldc + n] = val;
      }
    }
  }
}

__global__ __launch_bounds__(128) void gemm_wmma_f32_kernel(
    const float* __restrict__ A, const float* __restrict__ B, float* __restrict__ D,
    int K, int lda, int ldb, int ldc, int Mst, int Nst,
    size_t strideA, size_t strideB, size_t strideC,
    const float* __restrict__ bias) {
  __shared__ float lA[2][64 * LDA_S];   // [row][k] padded stride
  __shared__ float lB[2][16 * 64];      // [k][col]

  const int tid  = threadIdx.x;
  const int lane = tid & 31;
  const int wave = tid >> 5;                  // 0..3
  const int wm = wave >> 1, wn = wave & 1;    // 2x2 wave grid
  const int rowB = blockIdx.y * 64;
  const int colB = blockIdx.x * 64;
  const int b = blockIdx.z;
  A += (size_t)b * strideA;
  B += (size_t)b * strideB;
  D += (size_t)b * strideC;

  // --- staging descriptors: 2 A-chunks + 2 B-chunks (16B each) per thread ---
  const int ar0 = tid >> 2,         aq0 = (tid & 3) << 2;          // chunk tid
  const int ar1 = (tid + 128) >> 2, aq1 = ((tid + 128) & 3) << 2;  // chunk tid+128
  const float* gA0 = A + (size_t)(rowB + ar0) * lda + aq0;
  const float* gA1 = A + (size_t)(rowB + ar1) * lda + aq1;
  const int bk0 = tid >> 4,         bq0 = (tid & 15) << 2;
  const int bk1 = (tid + 128) >> 4, bq1 = ((tid + 128) & 15) << 2;
  const float* gB0 = B + (size_t)bk0 * ldb + colB + bq0;
  const float* gB1 = B + (size_t)bk1 * ldb + colB + bq1;

  const uint32_t lAd0 = (uint32_t)(uintptr_t)&lA[0][ar0 * LDA_S + aq0];
  const uint32_t lAd1 = (uint32_t)(uintptr_t)&lA[0][ar1 * LDA_S + aq1];
  const uint32_t lBd0 = (uint32_t)(uintptr_t)&lB[0][bk0 * 64 + bq0];
  const uint32_t lBd1 = (uint32_t)(uintptr_t)&lB[0][bk1 * 64 + bq1];
  const uint32_t bufA = sizeof(float) * 64 * LDA_S;
  const uint32_t bufB = sizeof(float) * 16 * 64;

  // --- per-lane compute indices (CDNA5 16x16x4 f32 WMMA layouts) ---
  const int khalf = (lane >> 4) << 1;         // 0 or 2
  const int mrow0 = wm * 32 + (lane & 15);
  const int mrow1 = mrow0 + 16;
  const int ncol0 = wn * 32 + (lane & 15);

  v8f acc00 = {}, acc01 = {}, acc10 = {}, acc11 = {};

  // prologue: stage slab 0 into buffer 0
  async_b128(lAd0, gA0);
  async_b128(lAd1, gA1);
  async_b128(lBd0, gB0);
  async_b128(lBd1, gB1);

  for (int k0 = 0, it = 0; k0 < K; k0 += 16, ++it) {
    const int sel = it & 1;
    if (k0 + 16 < K) {
      // stage next slab into the other buffer, then wait for current slab
      const uint32_t off = (sel ^ 1) ? 1u : 0u;
      async_b128(lAd0 + off * bufA, gA0 + (k0 + 16));
      async_b128(lAd1 + off * bufA, gA1 + (k0 + 16));
      async_b128(lBd0 + off * bufB, gB0 + (size_t)(k0 + 16) * ldb);
      async_b128(lBd1 + off * bufB, gB1 + (size_t)(k0 + 16) * ldb);
      asm volatile("s_wait_asynccnt 0x4" ::: "memory");
    } else {
      asm volatile("s_wait_asynccnt 0x0" ::: "memory");
    }
    __syncthreads();  // slab `sel` visible to all waves

    const float* sA = &lA[sel][0];
    const float* sB = &lB[sel][0];
#pragma unroll
    for (int kk = 0; kk < 16; kk += 4) {
      const int kx = kk + khalf;
      v2f a0 = *(const v2f*)(sA + mrow0 * LDA_S + kx);
      v2f a1 = *(const v2f*)(sA + mrow1 * LDA_S + kx);
      v2f b0, b1;
      b0.x = sB[kx * 64 + ncol0];
      b0.y = sB[(kx + 1) * 64 + ncol0];
      b1.x = sB[kx * 64 + ncol0 + 16];
      b1.y = sB[(kx + 1) * 64 + ncol0 + 16];
      acc00 = __builtin_amdgcn_wmma_f32_16x16x4_f32(false, a0, false, b0, (short)0, acc00, false, false);
      acc01 = __builtin_amdgcn_wmma_f32_16x16x4_f32(false, a0, false, b1, (short)0, acc01, false, false);
      acc10 = __builtin_amdgcn_wmma_f32_16x16x4_f32(false, a1, false, b0, (short)0, acc10, false, false);
      acc11 = __builtin_amdgcn_wmma_f32_16x16x4_f32(false, a1, false, b1, (short)0, acc11, false, false);
    }
    __syncthreads();  // everyone done reading `sel` before next stage overwrites
  }

  const int row0 = rowB + wm * 32;
  const int col0 = colB + wn * 32;
  store_c_frag(D, ldc, Mst, Nst, row0,      col0,      acc00, lane, bias);
  store_c_frag(D, ldc, Mst, Nst, row0,      col0 + 16, acc01, lane, bias);
  store_c_frag(D, ldc, Mst, Nst, row0 + 16, col0,      acc10, lane, bias);
  store_c_frag(D, ldc, Mst, Nst, row0 + 16, col0 + 16, acc11, lane, bias);
}

// ---------------------------------------------------------------------------
// Softmax over rows (axis O) per column c < 271; one wave32 per column.
// ---------------------------------------------------------------------------
__global__ __launch_bounds__(32) void softmax_col_kernel(
    float* __restrict__ att, int O, int ld) {
  const int c = blockIdx.x;
  const int lane = threadIdx.x;
  float m = -INFINITY;
  for (int s = lane; s < O; s += 32) m = fmaxf(m, att[(size_t)s * ld + c]);
#pragma unroll
  for (int off = 16; off >= 1; off >>= 1) m = fmaxf(m, __shfl_xor(m, off, 32));
  float sum = 0.f;
  for (int s = lane; s < O; s += 32) sum += expf(att[(size_t)s * ld + c] - m);
#pragma unroll
  for (int off = 16; off >= 1; off >>= 1) sum += __shfl_xor(sum, off, 32);
  const float inv = 1.f / sum;
  for (int s = lane; s < O; s += 32)
    att[(size_t)s * ld + c] = expf(att[(size_t)s * ld + c] - m) * inv;
}

// ---------------------------------------------------------------------------
extern "C" void kernel_launch(void* const* d_in, const int* in_sizes, int n_in,
                              void* d_out, int out_size, void* d_ws, size_t ws_size,
                              hipStream_t stream) {
  const float* x      = (const float*)d_in[0];  // [64,271,512]
  const float* ff     = (const float*)d_in[1];  // [512,32,32,4]
  const float* pos    = (const float*)d_in[2];  // [271,2]
  const float* proj_w = (const float*)d_in[3];  // [512,271]
  const float* proj_b = (const float*)d_in[4];  // [512]
  float* out = (float*)d_out;                   // [64,512,271]

  float* ws    = (float*)d_ws;
  float* Fmat  = ws;                              // 512*128
  float* Smat  = Fmat + ODIM * 128;               // 128*320 (cols >=271 zero)
  float* att   = Smat + 128 * CPAD;               // 512*320 (cols >=271 zero)
  float* Wp    = att + ODIM * CPAD;               // 512*272 (col 271 zero)
  float* T     = Wp + ODIM * KPW;                 // 64*320*320
  float* xp    = T + (size_t)BDIM * CPAD * CPAD;  // 64*320*512 (rows >=271 zero)

  // Prep / padding
  reduce_fourier_kernel<<<dim3(ODIM), dim3(128), 0, stream>>>(ff, Fmat);
  build_smat_kernel<<<dim3(CPAD), dim3(128), 0, stream>>>(pos, Smat);
  pad_projw_kernel<<<dim3((ODIM * KPW + 255) / 256), dim3(256), 0, stream>>>(proj_w, Wp);
  {
    const size_t total = (size_t)BDIM * CPAD * (SDIM / 4);
    pad_x_kernel<<<dim3((unsigned)((total + 255) / 256)), dim3(256), 0, stream>>>(x, xp);
  }

  // logits[512,320] = F[512,128] @ S[128,320]
  gemm_wmma_f32_kernel<<<dim3(CPAD / 64, ODIM / 64, 1), dim3(128), 0, stream>>>(
      Fmat, Smat, att, 128, 128, CPAD, CPAD, ODIM, CPAD, 0, 0, 0, nullptr);

  // softmax over O per column (columns sum to 1 -> proj_b passes through)
  softmax_col_kernel<<<dim3(CDIM), dim3(32), 0, stream>>>(att, ODIM, CPAD);

  // T[b][320,320] = xp[b][320,512] @ att[512,320] (padded rows/cols -> zeros)
  gemm_wmma_f32_kernel<<<dim3(CPAD / 64, CPAD / 64, BDIM), dim3(128), 0, stream>>>(
      xp, att, T, SDIM, SDIM, CPAD, CPAD, CPAD, CPAD,
      (size_t)CPAD * SDIM, 0, (size_t)CPAD * CPAD, nullptr);

  // out[b][512,271] = Wp[512,272] @ T[b][272,320] + proj_b  (row 271 of T = 0)
  gemm_wmma_f32_kernel<<<dim3(CPAD / 64, ODIM / 64, BDIM), dim3(128), 0, stream>>>(
      Wp, T, out, KPW, KPW, CPAD, CDIM, ODIM, CDIM,
      0, (size_t)CPAD * CPAD, (size_t)ODIM * CDIM, proj_b);
}